// ConvSurface_79757542686884
// MI455X (gfx1250) — compile-verified
//
#include <hip/hip_runtime.h>
#include <hip/hip_bf16.h>

// ConvSurface: per (mesh,face): 24 barycentric sample points on 3 neighbor
// triangles -> dir = point - center -> relu(dir . W^T + b) (64 ch) -> max over
// the 24 samples. Memory-bound (~53 MB touched vs ~1.2 GFLOP) -> f32 all the
// way; the tiny K=3 GEMM + bias rides V_WMMA_F32_16X16X4_F32 with K padded to
// 4 (dir -> [dx,dy,dz,1], W row -> [w0,w1,w2,bias]).
//
// One wave32 per face:
//   A tiles (16x4 f32, 2 VGPRs): rows = samples. Tile0 = s 0..15,
//   tile1 = s 8..23 (overlap harmless under max-pool).
//   B tiles (4x16 f32, 2 VGPRs): cols = 16 kernels; 4 column tiles -> 64.
//   8 WMMAs/wave. Sample-max: 16 in-lane v_max + one DS_SWIZZLE SWAPX16.
//   Output: two full-width 32-lane coalesced b32 stores (no predication).

#define NUM_MESHES 8
#define NUM_FACES 16384
#define NUM_SAMPLES 24
#define NUM_KERNEL 64

typedef __attribute__((ext_vector_type(2))) float v2f;
typedef __attribute__((ext_vector_type(8))) float v8f;

__global__ __launch_bounds__(256)
void conv_surface_wmma(const float* __restrict__ centers,   // (8,16384,3)
                       const float* __restrict__ nc,        // (8,16384,3,3,3)
                       const float* __restrict__ alpha,     // (16384,24)
                       const float* __restrict__ beta,      // (16384,24)
                       const float* __restrict__ gamma,     // (16384,24)
                       const float* __restrict__ W,         // (64,3)
                       const float* __restrict__ bias,      // (64,)
                       float* __restrict__ out)             // (8,16384,64)
{
    const int tid  = blockIdx.x * blockDim.x + threadIdx.x;
    const int wave = tid >> 5;                // one wave per (mesh,face)
    const int lane = threadIdx.x & 31;
    const int half = lane >> 4;               // 0: K=0,1 lanes   1: K=2,3 lanes
    const int r    = lane & 15;               // row (sample) / col (kernel) id

    const int m = wave >> 14;                 // / NUM_FACES
    const int f = wave & (NUM_FACES - 1);

    const size_t mf  = (size_t)m * NUM_FACES + f;
    const float* ctr = centers + mf * 3;
    const float cx = ctr[0], cy = ctr[1], cz = ctr[2];

    // ---- Build the two A tiles: A[M][K] with K in {dx,dy,dz,1} ----
    // lane<16 : VGPR0 = dx(sample), VGPR1 = dy(sample)
    // lane>=16: VGPR0 = dz(sample), VGPR1 = 1.0
    v2f A[2];
#pragma unroll
    for (int t = 0; t < 2; ++t) {
        const int s   = r + 8 * t;            // t=0: 0..15, t=1: 8..23
        const int nbr = s % 3;
        const float* crn = nc + (mf * 3 + (size_t)nbr) * 9;   // [corner j][xyz]
        const float al = alpha[f * NUM_SAMPLES + s];
        const float be = beta [f * NUM_SAMPLES + s];
        const float ga = gamma[f * NUM_SAMPLES + s];

        // component index via address math (no branches)
        const int   c0   = half ? 2  : 0;
        const float sub0 = half ? cz : cx;
        const float a0 = al * crn[c0] + be * crn[3 + c0] + ga * crn[6 + c0] - sub0;
        // load y-components unconditionally (same cachelines), select value
        const float a1v = al * crn[1] + be * crn[4] + ga * crn[7] - cy;
        A[t][0] = a0;
        A[t][1] = half ? 1.0f : a1v;
    }

    const v8f czero = {0.f, 0.f, 0.f, 0.f, 0.f, 0.f, 0.f, 0.f};
    float cm[4];

    // ---- 4 column tiles of 16 kernels each ----
#pragma unroll
    for (int kt = 0; kt < 4; ++kt) {
        const int col = kt * 16 + r;
        // B[K][N]: lane<16 holds K=0,1 ; lane>=16 holds K=2,3(=bias row).
        // Select the *address*, load once -> no EXEC branching.
        const float* pB0 = W + col * 3 + (half ? 2 : 0);
        const float* pB1 = half ? (bias + col) : (W + col * 3 + 1);
        v2f B;
        B[0] = *pB0;
        B[1] = *pB1;

        // D = A x B + 0 -> dir.W^T + b (bias folded into the K=3 row)
        v8f d0 = __builtin_amdgcn_wmma_f32_16x16x4_f32(
            false, A[0], false, B, (short)0, czero, false, false);
        v8f d1 = __builtin_amdgcn_wmma_f32_16x16x4_f32(
            false, A[1], false, B, (short)0, czero, false, false);

        // relu + max over samples:  max_s relu(x_s) == max(0, max_s x_s)
        float v = 0.0f;
#pragma unroll
        for (int i = 0; i < 8; ++i) {
            v = fmaxf(v, d0[i]);
            v = fmaxf(v, d1[i]);
        }
        // D layout: lane L holds rows 0..7 (L<16) or 8..15 (L>=16) of column
        // N=L%16. Combine halves with DS_SWIZZLE SWAPX16 (xor=0x10, and=0x1f).
        const int sw = __builtin_amdgcn_ds_swizzle(__float_as_int(v), 0x401f);
        cm[kt] = fmaxf(v, __int_as_float(sw));
    }

    // After the swap, lane r and lane 16+r both hold column kt*16+r.
    // => lane L holds col L via tile (L<16 ? 0 : 1), and col 32+L via
    //    tile (L<16 ? 2 : 3): two full 32-lane contiguous stores.
    float* outp = out + mf * NUM_KERNEL;
    outp[lane]      = half ? cm[1] : cm[0];   // cols  0..31, 128B coalesced
    outp[32 + lane] = half ? cm[3] : cm[2];   // cols 32..63, 128B coalesced
}

extern "C" void kernel_launch(void* const* d_in, const int* in_sizes, int n_in,
                              void* d_out, int out_size, void* d_ws, size_t ws_size,
                              hipStream_t stream) {
    const float* centers = (const float*)d_in[0];
    // d_in[1] = ring_n : unused by the reference
    const float* nc      = (const float*)d_in[2];
    const float* alpha   = (const float*)d_in[3];
    const float* beta    = (const float*)d_in[4];
    const float* gamma   = (const float*)d_in[5];
    const float* W       = (const float*)d_in[6];
    const float* b       = (const float*)d_in[7];
    float* out = (float*)d_out;

    const int total_waves = NUM_MESHES * NUM_FACES;       // 131072 (1/face)
    const int threads = 256;                              // 8 waves / block
    const int blocks  = total_waves * 32 / threads;       // 16384
    conv_surface_wmma<<<blocks, threads, 0, stream>>>(
        centers, nc, alpha, beta, gamma, W, b, out);
}